// OnnxPostprocess_63136019251696
// MI455X (gfx1250) — compile-verified
//
#include <hip/hip_runtime.h>
#include <stdint.h>

#define NUM_B 8
#define NUM_A 3
#define HWN   5776
#define NROW  (NUM_A * HWN)      // 17328 boxes per image
#define NUM_C 80
#define PAIRS (NUM_B * NUM_C)    // 640
#define TOPK  200
#define CAP   512
#define BINS  1024
#define NPAD  208                // 13 * 16
#define NTILE 13
#define SCORE_T 0.5f

typedef __attribute__((ext_vector_type(2))) float v2f;
typedef __attribute__((ext_vector_type(8))) float v8f;

// Map score bits (s in (0.5, 1.0) -> bits in (0x3F000000, 0x3F800000)) to 1024 bins,
// monotone in value. Clamped for robustness.
__device__ __forceinline__ uint32_t score_bin(uint32_t bits) {
    uint32_t t = bits >> 13;
    t = (t < 0x1F800u) ? 0u : (t - 0x1F800u);
    return t > 1023u ? 1023u : t;
}

// ---------------- K0: zero histograms ----------------
__global__ void k_zero(uint32_t* __restrict__ p, int n) {
    int i = blockIdx.x * blockDim.x + threadIdx.x;
    if (i < n) p[i] = 0u;
}

// ---------------- K1: per-(b,c) score histograms (coalesced float4 scan) -------------
__global__ void k_hist(const float4* __restrict__ sc4, uint32_t* __restrict__ hist, int n4) {
    int stride = gridDim.x * blockDim.x;
    for (int i = blockIdx.x * blockDim.x + threadIdx.x; i < n4; i += stride) {
        float4 v = sc4[i];
        int f = i * 4;
        int c = f % NUM_C;           // c is a multiple of 4; lanes j cover c..c+3
        int row = f / NUM_C;
        int b = row / NROW;
        int base = (b * NUM_C + c) << 10;
        float vv[4] = {v.x, v.y, v.z, v.w};
#pragma unroll
        for (int j = 0; j < 4; ++j) {
            if (vv[j] > SCORE_T) {
                uint32_t bin = score_bin(__float_as_uint(vv[j]));
                atomicAdd(&hist[base + j * BINS + bin], 1u);
            }
        }
    }
}

// ---------------- K2: per-pair cutoff bin (exact top-200 bound), zero counters -------
__global__ void k_cutoff(const uint32_t* __restrict__ hist,
                         uint32_t* __restrict__ cutoff, uint32_t* __restrict__ cnt) {
    __shared__ uint32_t h[BINS];
    int pair = blockIdx.x;
    for (int i = threadIdx.x; i < BINS; i += blockDim.x) h[i] = hist[(pair << 10) + i];
    __syncthreads();
    if (threadIdx.x == 0) {
        uint32_t acc = 0, cb = 0;
        for (int bin = BINS - 1; bin >= 0; --bin) {
            acc += h[bin];
            if (acc >= TOPK) { cb = (uint32_t)bin; break; }
        }
        cutoff[pair] = cb;   // if total < 200, cb==0 -> take everything (< 200 items)
        cnt[pair] = 0u;
    }
}

// ---------------- K3: collect survivor keys (score_bits<<32 | ~idx) ------------------
__global__ void k_collect(const float4* __restrict__ sc4, const uint32_t* __restrict__ cutoff,
                          uint32_t* __restrict__ cnt, uint64_t* __restrict__ cand, int n4) {
    __shared__ uint32_t s_cut[PAIRS];
    for (int i = threadIdx.x; i < PAIRS; i += blockDim.x) s_cut[i] = cutoff[i];
    __syncthreads();
    int stride = gridDim.x * blockDim.x;
    for (int i = blockIdx.x * blockDim.x + threadIdx.x; i < n4; i += stride) {
        float4 v = sc4[i];
        int f = i * 4;
        int c = f % NUM_C;
        int row = f / NUM_C;
        int b = row / NROW;
        int n = row - b * NROW;
        int pbase = b * NUM_C + c;
        float vv[4] = {v.x, v.y, v.z, v.w};
#pragma unroll
        for (int j = 0; j < 4; ++j) {
            if (vv[j] > SCORE_T) {
                uint32_t bits = __float_as_uint(vv[j]);
                if (score_bin(bits) >= s_cut[pbase + j]) {
                    uint32_t pos = atomicAdd(&cnt[pbase + j], 1u);
                    if (pos < CAP)
                        cand[(uint32_t)(pbase + j) * CAP + pos] =
                            ((uint64_t)bits << 32) | (uint32_t)(~(uint32_t)n);
                }
            }
        }
    }
}

// ---------------- K4: sort + WMMA-assisted IoU + greedy NMS + output -----------------
__global__ void __launch_bounds__(256) k_nms(const float4* __restrict__ boxes,
                                             const uint64_t* __restrict__ cand,
                                             const uint32_t* __restrict__ cnt,
                                             float* __restrict__ out) {
    __shared__ uint64_t s_key[CAP];
    __shared__ float4   s_box[NPAD];
    __shared__ float    s_area[NPAD];
    __shared__ uint32_t s_mask32[NPAD * 8];   // 208 rows x 16 halfwords (bitmask cols)
    __shared__ uint32_t s_keep[8];

    unsigned short* s_mask = (unsigned short*)s_mask32;
    int pair = blockIdx.x;
    int b = pair / NUM_C;
    int c = pair % NUM_C;
    int tid = threadIdx.x;
    int lane = tid & 31;
    int wave = tid >> 5;

    uint32_t m = cnt[pair];
    if (m > CAP) m = CAP;
    for (int i = tid; i < CAP; i += 256)
        s_key[i] = (i < (int)m) ? cand[(uint32_t)pair * CAP + i] : 0ull;
    __syncthreads();

    // Bitonic sort, descending (key = score desc, then index asc via ~idx).
    for (int k = 2; k <= CAP; k <<= 1) {
        for (int j = k >> 1; j > 0; j >>= 1) {
            for (int i = tid; i < CAP; i += 256) {
                int ix = i ^ j;
                if (ix > i) {
                    uint64_t a = s_key[i], bb = s_key[ix];
                    bool up = ((i & k) == 0);
                    if (up ? (a < bb) : (a > bb)) { s_key[i] = bb; s_key[ix] = a; }
                }
            }
            __syncthreads();
        }
    }

    // Gather boxes & areas for top-200 (invalid/padded -> zero box, zero area).
    for (int i = tid; i < NPAD; i += 256) {
        float4 bx = make_float4(0.f, 0.f, 0.f, 0.f);
        if (i < TOPK) {
            uint64_t key = s_key[i];
            if (key) {
                uint32_t n = ~(uint32_t)key;
                bx = boxes[(size_t)b * NROW + n];
            }
        }
        s_box[i]  = bx;
        s_area[i] = (bx.z - bx.x) * (bx.w - bx.y);
        s_mask[i * 16 + 13] = 0;   // pad halfword read by scan word 6
    }
    __syncthreads();

    // Suppression bit-matrix, upper-triangle tiles only (91 of 13x13).
    // area_i + area_j per 16x16 tile via two chained V_WMMA_F32_16X16X4_F32:
    //   D = Acol(area_i) x OnesRow;  D += OnesCol x Brow(area_j)
    bool lo  = lane < 16;
    int  col = lane & 15;
    for (int t = wave; t < 91; t += 8) {
        int ti = 0, rem = t;
        while (rem >= NTILE - ti) { rem -= NTILE - ti; ++ti; }
        int tj = ti + rem;

        v2f A1, B1, A2, B2;
        A1.x = lo ? s_area[ti * 16 + col] : 0.f;  A1.y = 0.f;
        B1.x = lo ? 1.f : 0.f;                    B1.y = 0.f;
        A2.x = lo ? 1.f : 0.f;                    A2.y = 0.f;
        B2.x = lo ? s_area[tj * 16 + col] : 0.f;  B2.y = 0.f;

        v8f acc = {};
        acc = __builtin_amdgcn_wmma_f32_16x16x4_f32(false, A1, false, B1,
                                                    (short)0, acc, false, false);
        acc = __builtin_amdgcn_wmma_f32_16x16x4_f32(false, A2, false, B2,
                                                    (short)0, acc, false, false);

        int nj = tj * 16 + col;
        float4 bn = s_box[nj];
#pragma unroll
        for (int r = 0; r < 8; ++r) {
            int mi = ti * 16 + (lo ? r : r + 8);
            float4 bm = s_box[mi];
            float xx1 = fmaxf(bm.x, bn.x), yy1 = fmaxf(bm.y, bn.y);
            float xx2 = fminf(bm.z, bn.z), yy2 = fminf(bm.w, bn.w);
            float inter = fmaxf(0.f, xx2 - xx1) * fmaxf(0.f, yy2 - yy1);
            float uni = acc[r] - inter;             // area_m + area_n - inter
            bool sup = (inter / uni) > 0.5f;        // NaN (0/0) compares false, as in ref
            uint32_t bal = __builtin_amdgcn_ballot_w32(sup);
            if (lane == 0) {
                s_mask[(ti * 16 + r    ) * 16 + tj] = (unsigned short)(bal & 0xFFFFu);
                s_mask[(ti * 16 + r + 8) * 16 + tj] = (unsigned short)(bal >> 16);
            }
        }
    }
    __syncthreads();

    // keep init: valid = (i < 200) && real candidate
    {
        bool valid = (tid < TOPK) && (s_key[tid] != 0ull);
        uint32_t bal = __builtin_amdgcn_ballot_w32(valid);
        if (lane == 0 && wave < 7) s_keep[wave] = bal;
    }
    __syncthreads();

    // Sequential greedy scan over 200 entries (7 words of 32 bits).
    if (tid == 0) {
        uint32_t kp[7];
#pragma unroll
        for (int w = 0; w < 7; ++w) kp[w] = s_keep[w];
        const uint32_t* m32 = (const uint32_t*)s_mask32;
        for (int i = 0; i < TOPK; ++i) {
            int w = i >> 5;
            if (!((kp[w] >> (i & 31)) & 1u)) continue;
            const uint32_t* row = m32 + i * 8;
            int bpos = i & 31;
            uint32_t low = (bpos == 31) ? 0xFFFFFFFFu : ((1u << (bpos + 1)) - 1u);
            kp[w] &= ~(row[w] & ~low);              // only j > i suppresses
            for (int ww = w + 1; ww < 7; ++ww) kp[ww] &= ~row[ww];
        }
#pragma unroll
        for (int w = 0; w < 7; ++w) s_keep[w] = kp[w];
    }
    __syncthreads();

    // Output (B, C, TOPK, 6): box4, score, class — zeros where not kept.
    if (tid < TOPK) {
        uint64_t key = s_key[tid];
        bool kept = (s_keep[tid >> 5] >> (tid & 31)) & 1u;
        float4 bx = s_box[tid];
        size_t base = ((size_t)pair * TOPK + tid) * 6;
        out[base + 0] = kept ? bx.x : 0.f;
        out[base + 1] = kept ? bx.y : 0.f;
        out[base + 2] = kept ? bx.z : 0.f;
        out[base + 3] = kept ? bx.w : 0.f;
        out[base + 4] = kept ? __uint_as_float((uint32_t)(key >> 32)) : 0.f;
        out[base + 5] = kept ? (float)c : 0.f;
    }
}

extern "C" void kernel_launch(void* const* d_in, const int* in_sizes, int n_in,
                              void* d_out, int out_size, void* d_ws, size_t ws_size,
                              hipStream_t stream) {
    const float4* boxes = (const float4*)d_in[0];   // (8, 17328, 4) f32
    const float4* sc4   = (const float4*)d_in[1];   // (8, 17328, 80) f32
    float* out = (float*)d_out;                     // (8, 80, 200, 6) f32

    uint8_t* ws = (uint8_t*)d_ws;
    uint32_t* hist   = (uint32_t*)ws;                                          // 2.62 MB
    uint64_t* cand   = (uint64_t*)(ws + (size_t)PAIRS * BINS * 4);             // 2.62 MB
    uint32_t* cnt    = (uint32_t*)(ws + (size_t)PAIRS * BINS * 4
                                      + (size_t)PAIRS * CAP * 8);              // 2.5 KB
    uint32_t* cutoff = cnt + PAIRS;                                            // 2.5 KB

    const int n4 = NUM_B * NROW * NUM_C / 4;  // 2,772,480 float4s

    k_zero<<<(PAIRS * BINS + 255) / 256, 256, 0, stream>>>(hist, PAIRS * BINS);
    k_hist<<<1280, 256, 0, stream>>>(sc4, hist, n4);
    k_cutoff<<<PAIRS, 256, 0, stream>>>(hist, cutoff, cnt);
    k_collect<<<1280, 256, 0, stream>>>(sc4, cutoff, cnt, cand, n4);
    k_nms<<<PAIRS, 256, 0, stream>>>(boxes, cand, cnt, out);
}